// MinDist_1408749273897
// MI455X (gfx1250) — compile-verified
//
#include <hip/hip_runtime.h>
#include <float.h>

typedef __attribute__((ext_vector_type(2))) float v2f;
typedef __attribute__((ext_vector_type(4))) float v4f;
typedef __attribute__((ext_vector_type(8))) float v8f;

#define NN 1024
#define DIM 128
#define QB 128          // queries per workgroup (8 waves x 16)
#define MB 64           // candidate rows per LDS chunk
#define LDS_STRIDE 132  // 128 + 4 pad: conflict-free b64 A-fragment reads
#define NCHUNK (NN / MB)

__global__ __launch_bounds__(256)
void mindist_wmma_async(const float* __restrict__ x,
                        const long long* __restrict__ batch,
                        float* __restrict__ out)
{
    __shared__ float bufs[2][MB * LDS_STRIDE];
    __shared__ float sqAll[NN];
    __shared__ int   bAll[NN];

    const int b    = blockIdx.y;
    const int q0   = blockIdx.x * QB;
    const int tid  = threadIdx.x;
    const int lane = tid & 31;
    const int wave = tid >> 5;
    const int hi   = lane >> 4;   // 0 -> K pair {0,1}, 1 -> K pair {2,3}
    const int lo   = lane & 15;

    const float* xb = x + (size_t)b * NN * DIM;

    // ---- async staging addresses (constant per thread) ----
    const int row = tid >> 2;                    // 0..63
    const int seg = tid & 3;                     // 32 floats per (row,seg)
    const unsigned voff = (unsigned)((row * DIM + seg * 32) * 4);          // global byte off in chunk
    const unsigned la0  = (unsigned)(uintptr_t)&bufs[0][row * LDS_STRIDE + seg * 32];
    const unsigned la1  = (unsigned)(uintptr_t)&bufs[1][row * LDS_STRIDE + seg * 32];

    // Issue one 64x128 chunk as 8 async b128 beats; IOFFSET applies to both
    // the global address and the LDS address, so one (vaddr, ldsaddr) pair
    // covers the whole contiguous 32-float span.
#define ISSUE_CHUNK(c)                                                         \
    do {                                                                       \
        const float* sb = xb + (size_t)(c) * MB * DIM;                         \
        unsigned la = ((c) & 1) ? la1 : la0;                                   \
        asm volatile(                                                          \
            "global_load_async_to_lds_b128 %0, %1, %2 offset:0\n\t"           \
            "global_load_async_to_lds_b128 %0, %1, %2 offset:16\n\t"          \
            "global_load_async_to_lds_b128 %0, %1, %2 offset:32\n\t"          \
            "global_load_async_to_lds_b128 %0, %1, %2 offset:48\n\t"          \
            "global_load_async_to_lds_b128 %0, %1, %2 offset:64\n\t"          \
            "global_load_async_to_lds_b128 %0, %1, %2 offset:80\n\t"          \
            "global_load_async_to_lds_b128 %0, %1, %2 offset:96\n\t"          \
            "global_load_async_to_lds_b128 %0, %1, %2 offset:112"             \
            :: "v"(la), "v"(voff), "s"(sb) : "memory");                        \
    } while (0)

    // Kick off the pipeline before doing any setup work.
    ISSUE_CHUNK(0);
    ISSUE_CHUNK(1);

    // ---- one-time setup: row norms + batch ids for all candidates ----
    for (int r = tid; r < NN; r += 256) {
        const float* xr = xb + (size_t)r * DIM;
        float s = 0.0f;
        #pragma unroll
        for (int j = 0; j < DIM / 4; ++j) {
            v4f v = *(const v4f*)(xr + 4 * j);
            s += v.x * v.x + v.y * v.y + v.z * v.z + v.w * v.w;
        }
        sqAll[r] = s;
        bAll[r]  = (int)batch[r];
    }

    // ---- B matrix (query^T, 128x16) held in registers for the whole kernel ----
    const int qn = q0 + wave * 16 + lo;          // this lane's query (column N = lo)
    const int bq = (int)batch[qn];
    v2f Bfrag[DIM / 4];
    {
        const float* xq = xb + (size_t)qn * DIM;
        #pragma unroll
        for (int kc = 0; kc < DIM / 4; ++kc)
            Bfrag[kc] = *(const v2f*)(xq + 4 * kc + 2 * hi);
    }

    float minval = FLT_MAX;
    int   minidx = 0;

    for (int c = 0; c < NCHUNK; ++c) {
        // chunk c's 8 async beats are complete when <=8*(chunks issued after c) remain
        if (c + 1 < NCHUNK) asm volatile("s_wait_asynccnt 8" ::: "memory");
        else                asm volatile("s_wait_asynccnt 0" ::: "memory");
        __syncthreads();                         // chunk c visible to all waves

        const float* buf = bufs[c & 1];
        const int    mc  = c * MB;

        #pragma unroll
        for (int s = 0; s < 4; ++s) {
            v8f acc = {};
            const float* arow = buf + (s * 16 + lo) * LDS_STRIDE + 2 * hi;
            #pragma unroll
            for (int kc = 0; kc < DIM / 4; ++kc) {
                v2f a = *(const v2f*)(arow + 4 * kc);       // ds_load b64
                acc = __builtin_amdgcn_wmma_f32_16x16x4_f32(
                        false, a, false, Bfrag[kc], (short)0, acc, false, false);
            }
            // acc[r] = G[cand = mc + s*16 + r + 8*hi][query = qn]
            #pragma unroll
            for (int r = 0; r < 8; ++r) {
                const int   mrow = s * 16 + r + 8 * hi;
                const int   cidx = mc + mrow;
                const float val  = sqAll[cidx] - 2.0f * acc[r];   // d2 - sq[n]
                const bool  ok   = (bAll[cidx] == bq) && (cidx != qn);
                if (ok && (val < minval || (val == minval && cidx < minidx))) {
                    minval = val;
                    minidx = cidx;
                }
            }
        }

        __syncthreads();                         // buffer (c&1) free for refill
        if (c + 2 < NCHUNK) ISSUE_CHUNK(c + 2);
    }

    // Lanes L and L^16 each hold 8 of the 16 rows of every tile: combine.
    {
        const float oval = __shfl_xor(minval, 16, 32);
        const int   oidx = __shfl_xor(minidx, 16, 32);
        if (oval < minval || (oval == minval && oidx < minidx)) {
            minval = oval; minidx = oidx;
        }
    }

    // out[b, q, :] = x[b,q,:] - x[b,nn(q),:]
    float* ob = out + (size_t)b * NN * DIM;
    for (int i = 0; i < 16; ++i) {
        const int nn_i = __shfl(minidx, i, 32);
        const int q_i  = q0 + wave * 16 + i;
        const float* xq = xb + (size_t)q_i * DIM + lane * 4;
        const float* xn = xb + (size_t)nn_i * DIM + lane * 4;
        float*       o  = ob + (size_t)q_i * DIM + lane * 4;
        v4f a = *(const v4f*)xq;
        v4f n = *(const v4f*)xn;
        *(v4f*)o = a - n;
    }
#undef ISSUE_CHUNK
}

extern "C" void kernel_launch(void* const* d_in, const int* in_sizes, int n_in,
                              void* d_out, int out_size, void* d_ws, size_t ws_size,
                              hipStream_t stream) {
    const float*     x     = (const float*)d_in[0];
    const long long* batch = (const long long*)d_in[1];
    float*           out   = (float*)d_out;
    const int B = in_sizes[0] / (NN * DIM);
    dim3 grid(NN / QB, B);
    mindist_wmma_async<<<grid, 256, 0, stream>>>(x, batch, out);
}